// DualLSTM_36155034698491
// MI455X (gfx1250) — compile-verified
//
#include <hip/hip_runtime.h>
#include <hip/hip_bf16.h>
#include <math.h>

// ---------------------------------------------------------------------------
// DualLSTM for MI455X (gfx1250, wave32).
//   V=32000, E=512, H=1024, S=2048, T=S-1=2047, Tpad=2048
// Pipeline:
//   1) init            : zero h/c state, y scratch, barrier counter, pad rows
//   2) embed_gather    : x[t] = embedding[sentence[t]]  (pad row zeroed)
//   3) gemm_wmma_f32   : gx_en = x @ W_ih_en^T          (v_wmma_f32_16x16x4_f32)
//   4) gemm_wmma_f32   : gx_cn = x @ W_ih_cn^T
//   5) scan_kernel     : persistent grid-cooperative LSTM scan (2 matvecs/step)
//   6) gemm_wmma_f32   : hid  = relu(outs @ fc_w1^T + b1)
//   7) gemm_wmma_f32   : out  = hid @ fc_w2^T + b2      -> d_out
//
// GEMM v2: block = 8 waves = 128M x 16N tile. The 16-col weight panel is
// shared by all 8 waves, staged in LDS via GLOBAL_LOAD_ASYNC_TO_LDS_B128
// (double-buffered, ASYNCcnt-synchronized) — 8x less weight traffic than
// per-wave global loads, and it exercises the CDNA5 async-to-LDS path.
// ---------------------------------------------------------------------------

typedef __attribute__((ext_vector_type(2))) float v2f;
typedef __attribute__((ext_vector_type(8))) float v8f;

#define T_LEN   2047
#define T_PAD   2048
#define E_DIM   512
#define H_DIM   1024
#define G_DIM   4096   // 4*H
#define V_DIM   32000
#define SCAN_WGS 32    // 32 * 256 = 8192 threads = rows of (yEn ++ yCn)

#define KC      32     // K-chunk staged per double-buffer slot
#define BSTRIDE 36     // LDS row stride (floats): 144B = 16B-aligned, bank-clean

// ---------------------------------------------------------------------------
// Stage one 16 x KC fp32 weight panel into LDS with async-to-LDS B128 loads.
// Threads 0..127 each move 16B:  thread i -> row n=i>>3, 16B-chunk q=i&7.
// ASYNCcnt-tracked; caller waits with s_wait_asynccnt + barrier.
// ---------------------------------------------------------------------------
__device__ inline void stage_panel_async(const float* __restrict__ W,
                                         int colBase, int kcBase, int K,
                                         float* buf)
{
    if (threadIdx.x < 128) {
        const int n = threadIdx.x >> 3;
        const int q = threadIdx.x & 7;
        const unsigned goff = (unsigned)((((size_t)(colBase + n) * K) + kcBase + q * 4) * 4u);
        const unsigned loff = (unsigned)(uintptr_t)(buf + n * BSTRIDE + q * 4);
        asm volatile("global_load_async_to_lds_b128 %0, %1, %2"
                     :: "v"(loff), "v"(goff), "s"(W) : "memory");
    }
}

__device__ inline void wait_async_all()
{
    asm volatile("s_wait_asynccnt 0x0" ::: "memory");
}

// ---------------------------------------------------------------------------
// fp32 WMMA GEMM:  C[M x N] = A[Mpad x K] @ W[N x K]^T (+bias)(+relu)
// Block: 256 threads / 8 waves -> 128M x 16N output tile.
// Wave w owns M-subtile (mg*128 + w*16); all waves share the LDS B panel.
// Full fp32 (V_WMMA_F32_16X16X4_F32): workload is HBM-bound, so the fp32
// matrix path costs nothing vs. bf16 and matches the reference bit-for-bit
// in accumulation precision.
// ---------------------------------------------------------------------------
__global__ void gemm_wmma_f32(const float* __restrict__ A,
                              const float* __restrict__ W,
                              const float* __restrict__ bias,
                              float* __restrict__ C,
                              int M, int N, int K, int relu)
{
    __shared__ float tileB[2][16 * BSTRIDE];     // double-buffered weight panel

    const int wave = threadIdx.x >> 5;           // 0..7  (wave-uniform)
    const int lane = threadIdx.x & 31;
    const int half = lane >> 4;                  // K-pair select (ISA 7.12.2)
    const int mr   = lane & 15;

    const int nTiles = N >> 4;
    const int tn     = blockIdx.x % nTiles;      // N-tile (shared by block)
    const int mg     = blockIdx.x / nTiles;      // 128-row M-group
    const int colBase = tn * 16;
    const int rowBase = mg * 128 + wave * 16;

    const float* __restrict__ arow = A + (size_t)(rowBase + mr) * K;

    // prime buffer 0
    stage_panel_async(W, colBase, 0, K, tileB[0]);

    v8f acc = {};
    int cur = 0;
    for (int kc = 0; kc < K; kc += KC, cur ^= 1) {
        wait_async_all();                        // my panel stores landed in LDS
        __syncthreads();                         // whole block sees the panel
        if (kc + KC < K) {                       // overlap: stream next panel
            stage_panel_async(W, colBase, kc + KC, K, tileB[cur ^ 1]);
            __builtin_prefetch(arow + kc + KC, 0, 1);   // global_prefetch_b8
        }

        const float* __restrict__ bp = tileB[cur];
#pragma unroll
        for (int kk = 0; kk < KC; kk += 4) {
            const int k0 = kk + 2 * half;
            const float2 av = *(const float2*)(arow + kc + k0);      // global
            const float2 bv = *(const float2*)(bp + mr * BSTRIDE + k0); // ds_load_b64
            v2f a; a.x = av.x; a.y = av.y;
            v2f b; b.x = bv.x; b.y = bv.y;
            acc = __builtin_amdgcn_wmma_f32_16x16x4_f32(
                      false, a, false, b, (short)0, acc, false, false);
        }
        __syncthreads();                         // done reading tileB[cur]
    }

    // D layout: VGPR r -> row rowBase + r + 8*half, col colBase + mr
    const int ocol = colBase + mr;
    const float bv = bias ? bias[ocol] : 0.0f;
#pragma unroll
    for (int r = 0; r < 8; ++r) {
        const int orow = rowBase + r + 8 * half;
        if (orow < M) {
            float v = acc[r] + bv;
            if (relu) v = v > 0.0f ? v : 0.0f;
            C[(size_t)orow * N + ocol] = v;
        }
    }
}

// ---------------------------------------------------------------------------
// Embedding gather; zero-fills the pad row t == T_LEN so GEMM A reads are safe.
// ---------------------------------------------------------------------------
__global__ void embed_gather(const int* __restrict__ sent,
                             const float* __restrict__ emb,
                             float* __restrict__ x)
{
    const int t = blockIdx.x;                       // 0..T_PAD-1
    const long src = (t < T_LEN) ? (long)sent[t] * E_DIM : -1;
    for (int e = threadIdx.x; e < E_DIM; e += blockDim.x)
        x[(long)t * E_DIM + e] = (src >= 0) ? emb[src + e] : 0.0f;
}

// ---------------------------------------------------------------------------
// Init: zero recurrent state, matvec scratch, grid-barrier counter, pad rows.
// ---------------------------------------------------------------------------
__global__ void init_state(float* __restrict__ h_en, float* __restrict__ h_cn,
                           float* __restrict__ c,
                           float* __restrict__ yEn, float* __restrict__ yCn,
                           unsigned* __restrict__ counter,
                           float* __restrict__ outs_pad,   // outs row T_LEN
                           float* __restrict__ hid_pad)    // hid  row T_LEN
{
    for (int i = threadIdx.x; i < H_DIM; i += blockDim.x) {
        h_en[i] = 0.0f; h_cn[i] = 0.0f; c[i] = 0.0f;
        outs_pad[i] = 0.0f; hid_pad[i] = 0.0f;
    }
    for (int i = threadIdx.x; i < G_DIM; i += blockDim.x) {
        yEn[i] = 0.0f; yCn[i] = 0.0f;
    }
    if (threadIdx.x < 16) counter[threadIdx.x] = 0u;
}

// ---------------------------------------------------------------------------
// Grid barrier: monotonic counter, target = epoch * gridDim.x.
// ---------------------------------------------------------------------------
__device__ inline void grid_sync(unsigned* cnt, unsigned target)
{
    __syncthreads();
    if (threadIdx.x == 0) {
        __threadfence();                         // publish our writes
        atomicAdd(cnt, 1u);
        while (__hip_atomic_load(cnt, __ATOMIC_RELAXED,
                                 __HIP_MEMORY_SCOPE_AGENT) < target) {
            __builtin_amdgcn_s_sleep(2);
        }
    }
    __syncthreads();
    __threadfence();                             // acquire others' writes
}

__device__ inline float sigf(float v) { return 1.0f / (1.0f + __expf(-v)); }

// ---------------------------------------------------------------------------
// Persistent grid-cooperative scan. 32 WGs x 256 threads = 8192 threads.
// Per step:
//   phase 1: thread j computes one row-dot:  j<4096 -> yEn[j] = W_hh_en[j].h_en
//                                            else   -> yCn[j-4096] = W_hh_cn[j-4096].h_cn
//            (h vectors staged in LDS; W_hh stays L2-resident: 33.6 MB << 192 MB)
//   phase 2: first 4 WGs update the 1024 hidden units (both branches share the
//            two matvecs — algebraic simplification of the reference's 4 cells)
// ---------------------------------------------------------------------------
__global__ void scan_kernel(const float* __restrict__ Whh_en,
                            const float* __restrict__ Whh_cn,
                            const int*   __restrict__ mask,
                            const float* __restrict__ gx_en,
                            const float* __restrict__ gx_cn,
                            float* __restrict__ h_en, float* __restrict__ h_cn,
                            float* __restrict__ c,
                            float* __restrict__ yEn, float* __restrict__ yCn,
                            float* __restrict__ outs,
                            unsigned* __restrict__ counter)
{
    __shared__ float sh[2 * H_DIM];              // h_en | h_cn
    const unsigned nwg = gridDim.x;
    unsigned epoch = 0;
    const int gtid = blockIdx.x * blockDim.x + threadIdx.x;   // 0..8191

    for (int t = 0; t < T_LEN; ++t) {
        // stage current hidden state into LDS
        for (int i = threadIdx.x; i < H_DIM; i += blockDim.x) {
            sh[i]          = h_en[i];
            sh[H_DIM + i]  = h_cn[i];
        }
        __syncthreads();

        // ---- phase 1: two shared matvecs --------------------------------
        {
            const bool en = gtid < G_DIM;
            const int  j  = en ? gtid : gtid - G_DIM;
            const float* __restrict__ wr = (en ? Whh_en : Whh_cn) + (long)j * H_DIM;
            const float* __restrict__ hv = en ? sh : (sh + H_DIM);
            float acc = 0.0f;
            for (int k = 0; k < H_DIM; k += 4) {
                const float4 w4 = *(const float4*)(wr + k);
                acc += w4.x * hv[k]     + w4.y * hv[k + 1]
                     + w4.z * hv[k + 2] + w4.w * hv[k + 3];
            }
            (en ? yEn : yCn)[j] = acc;
        }
        grid_sync(counter, (++epoch) * nwg);

        // ---- phase 2: elementwise cell updates (units 0..H-1) -----------
        if (gtid < H_DIM) {
            const int u = gtid;
            const float yEi = yEn[u], yEf = yEn[H_DIM + u];
            const float yEg = yEn[2 * H_DIM + u], yEo = yEn[3 * H_DIM + u];
            const float yCi = yCn[u], yCf = yCn[H_DIM + u];
            const float yCg = yCn[2 * H_DIM + u], yCo = yCn[3 * H_DIM + u];
            const long gb = (long)t * G_DIM;
            const float gei = gx_en[gb + u],            gef = gx_en[gb + H_DIM + u];
            const float geg = gx_en[gb + 2*H_DIM + u],  geo = gx_en[gb + 3*H_DIM + u];
            const float gci = gx_cn[gb + u],            gcf = gx_cn[gb + H_DIM + u];
            const float gcg = gx_cn[gb + 2*H_DIM + u],  gco = gx_cn[gb + 3*H_DIM + u];
            const float cc = c[u];

            // branch A (english token): cell(ge,h_en,c,W_en) then cell(0,h_cn,.,W_cn)
            const float cA1   = sigf(gef + yEf) * cc + sigf(gei + yEi) * tanhf(geg + yEg);
            const float hA_en = sigf(geo + yEo) * tanhf(cA1);
            const float cA2   = sigf(yCf) * cA1 + sigf(yCi) * tanhf(yCg);
            const float hA_cn = sigf(yCo) * tanhf(cA2);

            // branch B (chinese token): cell(gc,h_cn,c,W_cn) then cell(0,h_en,.,W_en)
            const float cB1   = sigf(gcf + yCf) * cc + sigf(gci + yCi) * tanhf(gcg + yCg);
            const float hB_cn = sigf(gco + yCo) * tanhf(cB1);
            const float cB2   = sigf(yEf) * cB1 + sigf(yEi) * tanhf(yEg);
            const float hB_en = sigf(yEo) * tanhf(cB2);

            const bool en = mask[t] > 0;
            const float he2 = en ? hA_en : hB_en;
            const float hc2 = en ? hA_cn : hB_cn;
            const float c2  = en ? cA2   : cB2;
            h_en[u] = he2; h_cn[u] = hc2; c[u] = c2;
            outs[(long)t * H_DIM + u] = he2 + hc2;
        }
        grid_sync(counter, (++epoch) * nwg);
    }
}

// ---------------------------------------------------------------------------
// Host-side launch
// ---------------------------------------------------------------------------
extern "C" void kernel_launch(void* const* d_in, const int* in_sizes, int n_in,
                              void* d_out, int out_size, void* d_ws, size_t ws_size,
                              hipStream_t stream)
{
    const int*   sentence = (const int*)  d_in[0];
    const int*   mask     = (const int*)  d_in[1];
    const float* emb      = (const float*)d_in[2];
    const float* W_ih_en  = (const float*)d_in[3];
    const float* W_hh_en  = (const float*)d_in[4];
    const float* W_ih_cn  = (const float*)d_in[5];
    const float* W_hh_cn  = (const float*)d_in[6];
    const float* fc_w1    = (const float*)d_in[7];
    const float* fc_b1    = (const float*)d_in[8];
    const float* fc_w2    = (const float*)d_in[9];
    const float* fc_b2    = (const float*)d_in[10];
    float* out = (float*)d_out;

    // workspace layout (floats)
    float* ws    = (float*)d_ws;
    float* x     = ws;                                       // Tpad x E
    float* gx_en = x     + (size_t)T_PAD * E_DIM;            // Tpad x 4H
    float* gx_cn = gx_en + (size_t)T_PAD * G_DIM;            // Tpad x 4H
    float* outs  = gx_cn + (size_t)T_PAD * G_DIM;            // Tpad x H
    float* hid   = outs  + (size_t)T_PAD * H_DIM;            // Tpad x H
    float* h_en  = hid   + (size_t)T_PAD * H_DIM;            // H
    float* h_cn  = h_en  + H_DIM;                            // H
    float* cst   = h_cn  + H_DIM;                            // H
    float* yEn   = cst   + H_DIM;                            // 4H
    float* yCn   = yEn   + G_DIM;                            // 4H
    unsigned* counter = (unsigned*)(yCn + G_DIM);            // 16

    // 1) init
    init_state<<<1, 256, 0, stream>>>(h_en, h_cn, cst, yEn, yCn, counter,
                                      outs + (size_t)T_LEN * H_DIM,
                                      hid  + (size_t)T_LEN * H_DIM);
    // 2) embedding gather (+ pad row zero)
    embed_gather<<<T_PAD, 256, 0, stream>>>(sentence, emb, x);

    // 3,4) input-gate GEMMs: [2048 x 512] @ [4096 x 512]^T
    {
        const int blocks = (T_PAD / 128) * (G_DIM / 16);     // 16 * 256 = 4096
        gemm_wmma_f32<<<blocks, 256, 0, stream>>>(x, W_ih_en, nullptr, gx_en,
                                                  T_LEN, G_DIM, E_DIM, 0);
        gemm_wmma_f32<<<blocks, 256, 0, stream>>>(x, W_ih_cn, nullptr, gx_cn,
                                                  T_LEN, G_DIM, E_DIM, 0);
    }

    // 5) recurrent scan (persistent, grid-cooperative)
    scan_kernel<<<SCAN_WGS, 256, 0, stream>>>(W_hh_en, W_hh_cn, mask,
                                              gx_en, gx_cn,
                                              h_en, h_cn, cst, yEn, yCn,
                                              outs, counter);

    // 6) hid = relu(outs @ fc_w1^T + b1): [2048 x 1024] @ [1024 x 1024]^T
    {
        const int blocks = (T_PAD / 128) * (H_DIM / 16);     // 16 * 64 = 1024
        gemm_wmma_f32<<<blocks, 256, 0, stream>>>(outs, fc_w1, fc_b1, hid,
                                                  T_LEN, H_DIM, H_DIM, 1);
    }

    // 7) out = hid @ fc_w2^T + b2: [2048 x 1024] @ [32000 x 1024]^T
    {
        const int blocks = (T_PAD / 128) * (V_DIM / 16);     // 16 * 2000 = 32000
        gemm_wmma_f32<<<blocks, 256, 0, stream>>>(hid, fc_w2, fc_b2, out,
                                                  T_LEN, V_DIM, H_DIM, 0);
    }
}